// CenterNet_31181462569049
// MI455X (gfx1250) — compile-verified
//
#include <hip/hip_runtime.h>

// CDNA5 / gfx1250 fused CenterNet loss.
// WMMA use: V_WMMA_F32_16X16X4_F32 computes 16x16 (location x box) squared-
// distance tiles as D = A(16x4) * B(4x16) + C, with A row = [g'x, g'y, 1, 0],
// B col = [-2c'x, -2c'y, |c'|^2, 0], C = |g'|^2, where g'/c' are coordinates
// translated by the tile's first grid point (translation-invariant dist2,
// keeps fp32 cancellation error small exactly where dist2 is small).
//
// Layout per ISA 7.12.2 (wave32):
//   A 16x4 f32  -> v2f: lanes 0-15 hold row M=lane, K={0,1};
//                  lanes 16-31 hold row M=lane-16, K={2,3}.
//   B 4x16 f32  -> v2f: lanes 0-15 col N=lane rows K={0,1};
//                  lanes 16-31 col N=lane-16 rows K={2,3}.
//   C/D 16x16   -> v8f: vgpr i, lanes 0-15 = (M=i, N=lane),
//                  lanes 16-31 = (M=i+8, N=lane-16).

typedef __attribute__((ext_vector_type(2))) float v2f;
typedef __attribute__((ext_vector_type(8))) float v8f;

#define INF_VAL   1.0e8f
#define M_TOT_C   87296
#define NBLK_MAIN 682      // 87296 / 128 locations per block

#if defined(__gfx1250__) && __has_builtin(__builtin_amdgcn_wmma_f32_16x16x4_f32)
#define USE_WMMA 1
#else
#define USE_WMMA 0
#endif

__global__ __launch_bounds__(256) void centernet_assign_loss(
    const float* __restrict__ boxes,   // [4,64,4]
    const float* __restrict__ agn,     // [87296]
    const float* __restrict__ regp,    // [87296,4]
    float* __restrict__ partials)      // [682,3] (reg_sum, reg_cnt, neg_sum)
{
    __shared__ float sb[8][256];       // x0,y0,x1,y1,cx,cy,inv_r2,crit per box
    __shared__ float red0[256], red1[256], red2[256];
    const int tid = threadIdx.x;

    // gfx1250 prefetch path (global_prefetch_b8) for this block's streams.
    __builtin_prefetch(&agn[blockIdx.x * 128 + (tid & 127)], 0, 1);
    __builtin_prefetch(&regp[blockIdx.x * 512 + tid * 2], 0, 1);

    // ---- per-box preprocessing: B*N = 256 boxes == blockDim.x ----
    {
        const float bx0 = boxes[tid * 4 + 0];
        const float by0 = boxes[tid * 4 + 1];
        const float bx1 = boxes[tid * 4 + 2];
        const float by1 = boxes[tid * 4 + 3];
        const float cx = 0.5f * (bx0 + bx1);
        const float cy = 0.5f * (by0 + by1);
        const float wb = bx1 - bx0, hb = by1 - by0;
        const float RK = (float)(((1.0 - 0.8) / (1.0 + 0.8)) *
                                 ((1.0 - 0.8) / (1.0 + 0.8)) * 2.0); // 2*DELTA^2
        const float r2 = fmaxf(RK * (wb * hb), 16.0f);
        sb[0][tid] = bx0; sb[1][tid] = by0; sb[2][tid] = bx1; sb[3][tid] = by1;
        sb[4][tid] = cx;  sb[5][tid] = cy;
        sb[6][tid] = 1.0f / r2;
        sb[7][tid] = sqrtf(wb * wb + hb * hb) * 0.5f;   // crit (per box)
    }
    __syncthreads();

    const int lane    = tid & 31;
    const int wv      = tid >> 5;
    const int j0      = blockIdx.x * 128 + wv * 16;     // first level-major index of tile
    const int halfsel = lane >> 4;

    // ---- level parameters (uniform per block; chunk sizes are multiples of 128) ----
    const int blk0 = blockIdx.x * 128;
    int s, lw, llc, lbase; float lo, hi;
    if      (blk0 < 65536) { s = 8;   lw = 7; llc = 14; lbase = 0;     lo = 0.0f;   hi = 80.0f;  }
    else if (blk0 < 81920) { s = 16;  lw = 6; llc = 12; lbase = 65536; lo = 64.0f;  hi = 160.0f; }
    else if (blk0 < 86016) { s = 32;  lw = 5; llc = 10; lbase = 81920; lo = 128.0f; hi = 320.0f; }
    else if (blk0 < 87040) { s = 64;  lw = 4; llc = 8;  lbase = 86016; lo = 256.0f; hi = 640.0f; }
    else                   { s = 128; lw = 3; llc = 6;  lbase = 87040; lo = 512.0f; hi = 1.0e7f; }
    const float fs = (float)s, half = 0.5f * fs, inv_s = 1.0f / fs;
    const int within  = j0 - lbase;
    const int b       = within >> llc;          // image index
    const int locBase = within & ((1 << llc) - 1);
    const int Wm      = (1 << lw) - 1;

    // tile-local translation origin = grid point of tile's first location
    const float G0x = (float)((locBase & Wm) * s) + half;
    const float G0y = (float)((locBase >> lw) * s) + half;

    // per-slot grid coords + |g'|^2 (C operand)
    float gxs[8], gys[8];
    v8f Cm;
#pragma unroll
    for (int i = 0; i < 8; ++i) {
        const int loc = locBase + i + (halfsel << 3);
        const float gx = (float)((loc & Wm) * s) + half;
        const float gy = (float)((loc >> lw) * s) + half;
        gxs[i] = gx; gys[i] = gy;
        const float dx = gx - G0x, dy = gy - G0y;     // exact (multiples of s)
        Cm[i] = dx * dx + dy * dy;
    }

#if USE_WMMA
    // A operand: row M = lane&15; lanes 0-15 carry K0,K1 = (g'x, g'y),
    // lanes 16-31 carry K2,K3 = (1, 0).
    v2f Am;
    {
        const int locA = locBase + (lane & 15);
        const float gax = (float)((locA & Wm) * s) + half - G0x;
        const float gay = (float)((locA >> lw) * s) + half - G0y;
        Am.x = halfsel ? 1.0f : gax;
        Am.y = halfsel ? 0.0f : gay;
    }
#endif

    float minAll[8], dmin[8]; int amin[8];
#pragma unroll
    for (int i = 0; i < 8; ++i) { minAll[i] = INF_VAL; dmin[i] = INF_VAL; amin[i] = 1 << 30; }

    const int boxb = b << 6;
#pragma unroll
    for (int bt = 0; bt < 4; ++bt) {
        const int myBox = (bt << 4) + (lane & 15);
        const int bi = boxb + myBox;
        const float x0 = sb[0][bi], y0 = sb[1][bi], x1 = sb[2][bi], y1 = sb[3][bi];
        const float cx = sb[4][bi], cy = sb[5][bi];
        const float ir2 = sb[6][bi], crit = sb[7][bi];

#if USE_WMMA
        const float cpx = cx - G0x, cpy = cy - G0y;
        v2f Bm;
        Bm.x = halfsel ? (cpx * cpx + cpy * cpy) : (-2.0f * cpx);
        Bm.y = halfsel ? 0.0f                   : (-2.0f * cpy);
        // D[m][n] = |g'|^2 - 2 g'.c' + |c'|^2 = dist2(loc m, box n)
        v8f D = __builtin_amdgcn_wmma_f32_16x16x4_f32(
            false, Am, false, Bm, (short)0, Cm, false, false);
#endif
        // per-box, per-level: discretized center and size-range mask
        const float cdx = truncf(cx * inv_s) * fs + half;
        const float cdy = truncf(cy * inv_s) * fs + half;
        const int cared = (crit >= lo) & (crit <= hi);   // branchless

#pragma unroll
        for (int i = 0; i < 8; ++i) {
            const float gx = gxs[i], gy = gys[i];
            const float l = gx - x0, t = gy - y0, r = x1 - gx, bo = y1 - gy;
            const float ml = fminf(fminf(l, t), fminf(r, bo));
            const float dx = gx - cdx, dy = gy - cdy;    // exact grid arithmetic
            // branchless predicates (avoid exec-save branch regions)
            const int is_in = (ml > 0.0f);
            const int peak  = (dx == 0.0f) & (dy == 0.0f);
            const int c3    = (fabsf(dx) <= fs) & (fabsf(dy) <= fs) & is_in;
#if USE_WMMA
            const float d2r = fmaxf(D[i], 0.0f);
#else
            const float ddx = gx - cx, ddy = gy - cy;
            const float d2r = ddx * ddx + ddy * ddy;
#endif
            const float d2 = peak ? 0.0f : d2r;
            const float w2 = d2 * ir2;
            minAll[i] = fminf(minAll[i], w2);
            const float dm = (c3 & cared) ? w2 : INF_VAL;
            const int take = (dm < dmin[i]);             // ascending box order => first index
            dmin[i] = take ? dm : dmin[i];
            amin[i] = take ? myBox : amin[i];
        }
    }

    // ---- reduce over the 16 boxes held across each 16-lane half ----
#pragma unroll
    for (int off = 8; off > 0; off >>= 1) {
#pragma unroll
        for (int i = 0; i < 8; ++i) {
            const float oA = __shfl_xor(minAll[i], off, 16);
            minAll[i] = fminf(minAll[i], oA);
            const float oD = __shfl_xor(dmin[i], off, 16);
            const int   oI = __shfl_xor(amin[i], off, 16);
            const int take = (oD < dmin[i]) | ((oD == dmin[i]) & (oI < amin[i]));
            dmin[i] = take ? oD : dmin[i];
            amin[i] = take ? oI : amin[i];
        }
    }

    // ---- per-location outputs: lanes 0-7 write locs 0-7, lanes 16-23 write locs 8-15 ----
    float regterm = 0.0f, regcnt = 0.0f, negterm = 0.0f;
    if ((lane & 15) < 8) {
        const int i = lane & 7;
        const int j = j0 + i + (halfsel << 3);
        const float gx = gxs[i], gy = gys[i];

        float hmv = expf(-minAll[i]);
        hmv = (hmv < 1.0e-4f) ? 0.0f : hmv;

        float pr = 1.0f / (1.0f + expf(-agn[j]));
        pr = fminf(fmaxf(pr, 1.0e-4f), 1.0f - 1.0e-4f);
        const float om = 1.0f - hmv;
        const float om2 = om * om;
        negterm = (pr < 0.85f) ? logf(1.0f - pr) * pr * pr * (om2 * om2) : 0.0f;

        if (dmin[i] < INF_VAL) {
            const int bi = boxb + amin[i];
            const float x0 = sb[0][bi], y0 = sb[1][bi], x1 = sb[2][bi], y1 = sb[3][bi];
            const float tl = (gx - x0) * inv_s, tt = (gy - y0) * inv_s;
            const float tr = (x1 - gx) * inv_s, tb = (y1 - gy) * inv_s;
            const float pl = regp[4 * j + 0], pt = regp[4 * j + 1];
            const float pr2 = regp[4 * j + 2], pb = regp[4 * j + 3];
            const float t_area = (tl + tr) * (tt + tb);
            const float p_area = (pl + pr2) * (pt + pb);
            const float w_i = fminf(pl, tl) + fminf(pr2, tr);
            const float g_w = fmaxf(pl, tl) + fmaxf(pr2, tr);
            const float h_i = fminf(pb, tb) + fminf(pt, tt);
            const float g_h = fmaxf(pb, tb) + fmaxf(pt, tt);
            const float ac = g_w * g_h + 1.0e-7f;
            const float inter = w_i * h_i;
            const float uni = t_area + p_area - inter;
            const float ious = (inter + 1.0f) / (uni + 1.0f);
            const float gious = ious - (ac - uni) / ac;
            regterm = 1.0f - gious;
            regcnt = 1.0f;
        }
    }

    // ---- deterministic block tree reduction ----
    red0[tid] = regterm; red1[tid] = regcnt; red2[tid] = negterm;
    __syncthreads();
    for (int st = 128; st > 0; st >>= 1) {
        if (tid < st) {
            red0[tid] += red0[tid + st];
            red1[tid] += red1[tid + st];
            red2[tid] += red2[tid + st];
        }
        __syncthreads();
    }
    if (tid == 0) {
        partials[blockIdx.x * 3 + 0] = red0[0];
        partials[blockIdx.x * 3 + 1] = red1[0];
        partials[blockIdx.x * 3 + 2] = red2[0];
    }
}

__global__ __launch_bounds__(256) void centernet_finalize(
    const float* __restrict__ boxes,
    const float* __restrict__ agn,
    const float* __restrict__ partials,
    float* __restrict__ out)
{
    __shared__ float r0[256], r1[256], r2[256], r3[256], r4[256];
    const int tid = threadIdx.x;

    float s0 = 0.0f, s1 = 0.0f, s2 = 0.0f;
    for (int p = tid; p < NBLK_MAIN; p += 256) {
        s0 += partials[3 * p + 0];
        s1 += partials[3 * p + 1];
        s2 += partials[3 * p + 2];
    }

    // positive-index focal terms: B*N*L = 4*64*5 = 1280 entries, 5 per thread
    const float strid[5] = {8.0f, 16.0f, 32.0f, 64.0f, 128.0f};
    const int   basesA[5] = {0, 65536, 81920, 86016, 87040};
    const int   locA[5]   = {16384, 4096, 1024, 256, 64};
    const int   wA[5]     = {128, 64, 32, 16, 8};
    const float loA[5]    = {0.0f, 64.0f, 128.0f, 256.0f, 512.0f};
    const float hiA[5]    = {80.0f, 160.0f, 320.0f, 640.0f, 1.0e7f};

    float ps = 0.0f, pc = 0.0f;
    for (int k = 0; k < 5; ++k) {
        const int e = tid * 5 + k;            // e = b*320 + n*5 + lvl
        const int bb = e / 320;
        const int rem = e - bb * 320;
        const int n = rem / 5;
        const int lvl = rem - n * 5;
        const float* bx = boxes + (bb * 64 + n) * 4;
        const float x0 = bx[0], y0 = bx[1], x1 = bx[2], y1 = bx[3];
        const float cx = 0.5f * (x0 + x1), cy = 0.5f * (y0 + y1);
        const int cix = (int)(cx / strid[lvl]);
        const int ciy = (int)(cy / strid[lvl]);
        int pos = basesA[lvl] + bb * locA[lvl] + ciy * wA[lvl] + cix;
        const float wb = x1 - x0, hb = y1 - y0;
        const float crit = sqrtf(wb * wb + hb * hb) * 0.5f;
        const bool mask = (crit >= loA[lvl]) && (crit <= hiA[lvl]);
        if (pos < 0) pos = 0;
        if (pos > M_TOT_C - 1) pos = M_TOT_C - 1;
        float pr = 1.0f / (1.0f + expf(-agn[pos]));
        pr = fminf(fmaxf(pr, 1.0e-4f), 1.0f - 1.0e-4f);
        if (mask) {
            const float o = 1.0f - pr;
            ps += logf(pr) * o * o;
            pc += 1.0f;
        }
    }

    r0[tid] = s0; r1[tid] = s1; r2[tid] = s2; r3[tid] = ps; r4[tid] = pc;
    __syncthreads();
    for (int st = 128; st > 0; st >>= 1) {
        if (tid < st) {
            r0[tid] += r0[tid + st];
            r1[tid] += r1[tid + st];
            r2[tid] += r2[tid + st];
            r3[tid] += r3[tid + st];
            r4[tid] += r4[tid + st];
        }
        __syncthreads();
    }
    if (tid == 0) {
        const float regsum = r0[0];
        const float regnorm = fmaxf(r1[0], 1.0f);
        const float negsum = r2[0];
        const float possum = r3[0];
        const float nposavg = fmaxf(r4[0], 1.0f);
        out[0] = 1.0f * regsum / regnorm;                        // REG_WEIGHT * ...
        out[1] = 0.5f * (0.25f * (-possum)) / nposavg;           // POS_W * ALPHA * ...
        out[2] = 0.5f * ((1.0f - 0.25f) * (-negsum)) / nposavg;  // NEG_W * (1-ALPHA) * ...
    }
}

extern "C" void kernel_launch(void* const* d_in, const int* in_sizes, int n_in,
                              void* d_out, int out_size, void* d_ws, size_t ws_size,
                              hipStream_t stream) {
    const float* boxes = (const float*)d_in[0];
    // d_in[1] = gt_classes (unused by the reference loss)
    const float* agn   = (const float*)d_in[2];
    const float* regp  = (const float*)d_in[3];
    float* out = (float*)d_out;
    float* partials = (float*)d_ws;   // 682*3 floats

    centernet_assign_loss<<<dim3(NBLK_MAIN), dim3(256), 0, stream>>>(boxes, agn, regp, partials);
    centernet_finalize<<<dim3(1), dim3(256), 0, stream>>>(boxes, agn, partials, out);
}